// DeformConv2D_60868276519450
// MI455X (gfx1250) — compile-verified
//
#include <hip/hip_runtime.h>
#include <hip/hip_bf16.h>

typedef float v2f __attribute__((ext_vector_type(2)));
typedef float v8f __attribute__((ext_vector_type(8)));

#define HW     16384        // 128*128
#define C_IN   64
#define C_OUT  128
#define KK     576          // C_IN * 9
#define NPIX   16           // pixels per workgroup tile
#define BS_PAD 580          // 576 + 4 pad -> conflict-free ds_load_b64

// ---------------------------------------------------------------------------
// Kernel 1: offset conv  (Conv2d 64 -> 18, 3x3, pad 1, stride 1) + bias
// out layout: off[b][18][128][128]
// ---------------------------------------------------------------------------
__global__ void __launch_bounds__(256)
offset_conv_kernel(const float* __restrict__ x,
                   const float* __restrict__ w_off,
                   const float* __restrict__ b_off,
                   float* __restrict__ off) {
    int idx = blockIdx.x * 256 + threadIdx.x;          // 4*18*128*128 = 1179648
    if (idx >= 4 * 18 * HW) return;
    int w  = idx & 127;
    int h  = (idx >> 7) & 127;
    int oc = (idx >> 14) % 18;
    int b  = idx / (18 * HW);

    float acc = b_off[oc];
    const float* xb = x + b * C_IN * HW;
    const float* wk = w_off + oc * C_IN * 9;

    for (int ci = 0; ci < C_IN; ++ci) {
        const float* xc = xb + ci * HW;
        const float* wc = wk + ci * 9;
        #pragma unroll
        for (int ky = 0; ky < 3; ++ky) {
            int yy = h + ky - 1;
            if (yy < 0 || yy > 127) continue;
            #pragma unroll
            for (int kx = 0; kx < 3; ++kx) {
                int xx = w + kx - 1;
                if (xx < 0 || xx > 127) continue;
                acc = fmaf(wc[ky * 3 + kx], xc[yy * 128 + xx], acc);
            }
        }
    }
    off[idx] = acc;
}

// ---------------------------------------------------------------------------
// Kernel 2: fused bilinear gather + WMMA GEMM
//   D[128 co, pix] = W[128, 576] x Xoff[576, pix]
// One workgroup = 16 pixels (same b,h; 16 consecutive w), 8 waves x 16 co each.
// ---------------------------------------------------------------------------
__global__ void __launch_bounds__(256)
deform_main_kernel(const float* __restrict__ x,      // [4][64][128][128]
                   const float* __restrict__ off,    // [4][18][128][128]
                   const float* __restrict__ wA,     // [128][576]
                   float* __restrict__ out) {        // [4][128][128][128]
    __shared__ float  Bs[NPIX][BS_PAD];   // gathered B tile [pix][k]
    __shared__ float4 gw[NPIX][9];        // bilinear weights (lt, rb, lb, rt)
    __shared__ int4   gidx[NPIX][9];      // flat x-indices, -1 == zero-pad

    const int t  = threadIdx.x;
    const int wg = blockIdx.x;                 // 4096 = 4 * 128 * 8
    const int b  = wg >> 10;
    const int h  = (wg & 1023) >> 3;
    const int w0 = (wg & 7) << 4;

    // ---- Phase 1: sampling geometry for 16 pixels x 9 taps ----------------
    if (t < NPIX * 9) {
        const int pix = t & 15;
        const int n   = t >> 4;                // 0..8
        const int w   = w0 + pix;

        const float off_y = off[(((b * 18 + n)      << 7) + h << 7) + w];
        const float off_x = off[(((b * 18 + 9 + n)  << 7) + h << 7) + w];

        // p = p0 + p_n + offset  (1-based coords in 130x130 padded image)
        const float py = (float)(h + 1) + (float)(n / 3 - 1) + off_y;
        const float px = (float)(w + 1) + (float)(n % 3 - 1) + off_x;

        const float fy = floorf(py), fx = floorf(px);
        int qly = (int)fy;       qly = min(max(qly, 0), 129);
        int qlx = (int)fx;       qlx = min(max(qlx, 0), 129);
        int qry = (int)fy + 1;   qry = min(max(qry, 0), 129);
        int qrx = (int)fx + 1;   qrx = min(max(qrx, 0), 129);

        // padding-region snap + clamp (pad=1 -> valid band [1,128])
        float psy = (py < 1.0f || py > 128.0f) ? fy : py;
        float psx = (px < 1.0f || px > 128.0f) ? fx : px;
        psy = fminf(fmaxf(psy, 0.0f), 129.0f);
        psx = fminf(fmaxf(psx, 0.0f), 129.0f);

        const float gl_y = 1.0f + ((float)qly - psy);
        const float gr_y = 1.0f - ((float)qry - psy);
        const float gl_x = 1.0f + ((float)qlx - psx);
        const float gr_x = 1.0f - ((float)qrx - psx);

        gw[pix][n] = make_float4(gl_y * gl_x,   // lt
                                 gr_y * gr_x,   // rb
                                 gl_y * gr_x,   // lb  (lt.y, rb.x)
                                 gr_y * gl_x);  // rt  (rb.y, lt.x)

        // resolve padded coords -> flat index into unpadded x, or -1 (zero)
        #define RESOLVE(qy, qx) \
            (((qy) >= 1 && (qy) <= 128 && (qx) >= 1 && (qx) <= 128) \
                 ? (((qy) - 1) * 128 + ((qx) - 1)) : -1)
        gidx[pix][n] = make_int4(RESOLVE(qly, qlx), RESOLVE(qry, qrx),
                                 RESOLVE(qly, qrx), RESOLVE(qry, qlx));
        #undef RESOLVE
    }
    __syncthreads();

    // ---- Phase 2: gather B tile: Bs[pix][ci*9 + n] ------------------------
    {
        const float* xb = x + b * C_IN * HW;
        // e = ci*144 + n*16 + pix  (pix fastest -> near-coalesced gathers)
        for (int e = t; e < C_IN * 144; e += 256) {
            const int pix = e & 15;
            const int rn  = e >> 4;
            const int n   = rn % 9;
            const int ci  = rn / 9;
            const float* xc = xb + ci * HW;

            const int4   id = gidx[pix][n];
            const float4 g  = gw[pix][n];
            float acc;
            acc  = g.x * (id.x >= 0 ? xc[id.x] : 0.0f);
            acc  = fmaf(g.y, (id.y >= 0 ? xc[id.y] : 0.0f), acc);
            acc  = fmaf(g.z, (id.z >= 0 ? xc[id.z] : 0.0f), acc);
            acc  = fmaf(g.w, (id.w >= 0 ? xc[id.w] : 0.0f), acc);
            Bs[pix][ci * 9 + n] = acc;
        }
    }
    __syncthreads();

    // ---- Phase 3: WMMA GEMM, wave w -> co block [16w, 16w+16) -------------
    const int lane = t & 31;
    const int wave = t >> 5;
    const int row  = lane & 15;       // A: M index / B: column (pixel)
    const int half = lane >> 4;       // K-group select

    // A lane layout: a[v] = W[co][k0 + 2*half + v]  (contiguous -> b64 load)
    const float* Arow = wA + (wave * 16 + row) * KK + 2 * half;
    // B lane layout: b[v] = Bs[row][k0 + 2*half + v] (contiguous -> ds b64)
    const float* Brow = &Bs[row][2 * half];

    v8f acc = {};
    #pragma unroll 4
    for (int k0 = 0; k0 < KK; k0 += 4) {
        v2f a = *(const v2f*)(Arow + k0);
        v2f bb = *(const v2f*)(Brow + k0);
        acc = __builtin_amdgcn_wmma_f32_16x16x4_f32(
            false, a, false, bb, (short)0, acc, false, false);
    }

    // ---- Store: D vgpr v -> co = wave*16 + v + 8*half, col = lane&15 ------
    const int wcol = w0 + row;
    #pragma unroll
    for (int v = 0; v < 8; ++v) {
        const int co = wave * 16 + v + 8 * half;
        out[((b * C_OUT + co) * 128 + h) * 128 + wcol] = acc[v];
    }
}

// ---------------------------------------------------------------------------
extern "C" void kernel_launch(void* const* d_in, const int* in_sizes, int n_in,
                              void* d_out, int out_size, void* d_ws, size_t ws_size,
                              hipStream_t stream) {
    const float* x      = (const float*)d_in[0];   // (4,64,128,128)
    const float* w_off  = (const float*)d_in[1];   // (18,64,3,3)
    const float* b_off  = (const float*)d_in[2];   // (18,)
    const float* w_conv = (const float*)d_in[3];   // (128,64,3,3) == [128][576]
    float* out = (float*)d_out;                    // (4,128,128,128)
    float* off = (float*)d_ws;                     // (4,18,128,128) = 4.7 MB

    offset_conv_kernel<<<(4 * 18 * HW + 255) / 256, 256, 0, stream>>>(
        x, w_off, b_off, off);

    deform_main_kernel<<<4 * 128 * 8, 256, 0, stream>>>(
        x, off, w_conv, out);
}